// LinearBucketEncoder_82145544503310
// MI455X (gfx1250) — compile-verified
//
#include <hip/hip_runtime.h>

typedef float v2f __attribute__((ext_vector_type(2)));
typedef float v8f __attribute__((ext_vector_type(8)));

#define NUM_COLS     96
#define OUT_CH       16
#define NUM_BUCKETS  4
#define ROWS_PER_WAVE   16
#define WAVES_PER_BLOCK 8
#define ROWS_PER_BLOCK  (ROWS_PER_WAVE * WAVES_PER_BLOCK)   // 128

// out[n, c, o] = sum_k enc(x[n,c], boundaries[c])_k * weight[c,k,o] + bias[c,o]
// One V_WMMA_F32_16X16X4_F32 per (16-row tile, column): A = enc (16x4),
// B = weight[c] (4x16), C preloaded with bias -> D stored directly (NT stores).
__global__ __launch_bounds__(256)
void linear_bucket_encoder_wmma(const float* __restrict__ x,
                                const float* __restrict__ boundaries,
                                const float* __restrict__ weight,
                                const float* __restrict__ bias,
                                float* __restrict__ out,
                                int n_rows) {
    const int lane = threadIdx.x & 31;
    const int wave = threadIdx.x >> 5;
    const int h    = lane >> 4;        // lane half: selects k-pair (0,1) or (2,3)
    const int n    = lane & 15;        // A: row-in-tile; B/D: output channel
    const int rowBase = blockIdx.x * ROWS_PER_BLOCK + wave * ROWS_PER_WAVE;
    if (rowBase >= n_rows) return;     // wave-uniform; EXEC stays all-ones past here

    const int k0 = 2 * h;              // this lane's first bucket index
    const int k1 = 2 * h + 1;

#pragma unroll 2
    for (int c = 0; c < NUM_COLS; ++c) {
        // ---- column-uniform parameters (scalarized to s_load) ----
        const float* bnd = boundaries + c * (NUM_BUCKETS + 1);
        const float b0 = bnd[0], b1 = bnd[1], b2 = bnd[2], b3 = bnd[3], b4 = bnd[4];

        // ---- per-lane x value: row = rowBase + (lane & 15) ----
        const float xv = x[(long)(rowBase + n) * NUM_COLS + c];

        // bucket index = #(x > inner), inner = boundaries[1..3]
        const int bidx = (xv > b1) + (xv > b2) + (xv > b3);

        // low/high boundary pair for this lane's two k values
        const float low0  = h ? b2 : b0;
        const float high0 = h ? b3 : b1;
        const float low1  = h ? b3 : b1;
        const float high1 = h ? b4 : b2;

        // At most one of this lane's two k values equals bidx -> ONE division,
        // executed unconditionally by every lane (no EXEC divergence).
        const bool m0 = (bidx == k0);
        const bool m1 = (bidx == k1);
        const float lowm  = m0 ? low0  : low1;
        const float highm = m0 ? high0 : high1;
        const float lin = (xv - lowm) / (highm - lowm + 1e-9f);

        // enc_k = one_hot_k * (x-low)/interval + (x>low) * (1 - one_hot_k)
        v2f a;
        a.x = m0 ? lin : ((xv > low0) ? 1.0f : 0.0f);
        a.y = m1 ? lin : ((xv > low1) ? 1.0f : 0.0f);

        // ---- B matrix: weight[c, k, n];  VGPR j / half h -> K = 2h + j ----
        const float* wcol = weight + c * (NUM_BUCKETS * OUT_CH);
        v2f b;
        b.x = wcol[k0 * OUT_CH + n];
        b.y = wcol[k1 * OUT_CH + n];

        // ---- C matrix: bias[c, n] broadcast into all 8 accumulator rows ----
        const float bv = bias[c * OUT_CH + n];
        v8f acc = { bv, bv, bv, bv, bv, bv, bv, bv };

        // D = A x B + C   (16x16 f32 tile: 16 rows x 16 out-channels)
        v8f d = __builtin_amdgcn_wmma_f32_16x16x4_f32(
            /*neg_a=*/false, a, /*neg_b=*/false, b,
            /*c_mod=*/(short)0, acc, /*reuse_a=*/false, /*reuse_b=*/false);

        // ---- store: VGPR j -> row rowBase + 8h + j, channel n ----
        // lanes 0..15 / 16..31 each write one contiguous 64B run per j.
        // Output is a pure write-once stream -> non-temporal stores.
        const long colBase = (long)c * OUT_CH + n;
        float* orow = out + (long)(rowBase + 8 * h) * (NUM_COLS * OUT_CH) + colBase;
#pragma unroll
        for (int j = 0; j < 8; ++j) {
            __builtin_nontemporal_store(d[j], orow);
            orow += NUM_COLS * OUT_CH;
        }
    }
}

extern "C" void kernel_launch(void* const* d_in, const int* in_sizes, int n_in,
                              void* d_out, int out_size, void* d_ws, size_t ws_size,
                              hipStream_t stream) {
    const float* x          = (const float*)d_in[0];
    const float* boundaries = (const float*)d_in[1];
    const float* weight     = (const float*)d_in[2];
    const float* bias       = (const float*)d_in[3];
    float* out              = (float*)d_out;

    const int n_rows = in_sizes[0] / NUM_COLS;          // 131072
    const int blocks = (n_rows + ROWS_PER_BLOCK - 1) / ROWS_PER_BLOCK;  // 1024

    linear_bucket_encoder_wmma<<<blocks, 256, 0, stream>>>(
        x, boundaries, weight, bias, out, n_rows);
}